// CARNN_fixed_28750511079472
// MI455X (gfx1250) — compile-verified
//
#include <hip/hip_runtime.h>

typedef float v2f __attribute__((ext_vector_type(2)));
typedef float v4f __attribute__((ext_vector_type(4)));
typedef float v8f __attribute__((ext_vector_type(8)));

#define D_MODEL    64
#define SEQ_LEN    9
#define ACTION_NUM 300
#define NT_OUT     19      // ceil(300/16)

// Packed-fragment workspace layout (floats):
//   wsM : [9][4 nt][8 ktp][32 lane][4]  -> 36864 floats at offset 0
//   wsW : same                          -> 36864 floats at offset 36864
//   wsO : [19 nt][8 ktp][32 lane][4]    -> 19456 floats at offset 73728
// Per lane, one b128 = fragments for kt = 2*ktp and 2*ktp+1 of one n-tile.
#define WS_M_OFF   0
#define WS_W_OFF   36864
#define WS_O_OFF   73728
#define WS_FLOATS  93184
#define WS_BYTES   (WS_FLOATS * 4)

__device__ __forceinline__ float sigmoid_f(float x) {
    return 1.0f / (1.0f + __expf(-x));
}

// ---------------- pre-pass: pack weights into WMMA B-fragment order ----------------
// B-frag (4x16 f32): lane gives column n = nt*16 + (lane&15); VGPR pair gives
// K = 4*kt + 2*(lane>>4) .. +1  => source value W[n][k], contiguous per pair.
__global__ void carnn_pack_weights(const float* __restrict__ Mw,   // [50][64][64]
                                   const float* __restrict__ Ww,   // [56][64][64]
                                   const float* __restrict__ outw, // [300][64]
                                   float* __restrict__ ws)
{
    const int idx = blockIdx.x * 256 + threadIdx.x;
    if (idx >= WS_FLOATS) return;
    float val;
    if (idx < WS_O_OFF) {
        const int matsel = idx / WS_W_OFF;      // 0 = M, 1 = W
        int r  = idx % WS_W_OFF;
        const int s   = r / 4096;  r %= 4096;
        const int nt  = r / 1024;  r %= 1024;
        const int ktp = r / 128;   r %= 128;
        const int lane = r / 4;
        const int j    = r % 4;
        const int h = lane >> 4, lr = lane & 15;
        const int n  = nt * 16 + lr;
        const int kt = 2 * ktp + (j >> 1);
        const int k  = 4 * kt + 2 * h + (j & 1);
        const float* src = matsel ? Ww : Mw;
        val = src[s * 4096 + n * D_MODEL + k];
    } else {
        int r = idx - WS_O_OFF;
        const int nt  = r / 1024;  r %= 1024;
        const int ktp = r / 128;   r %= 128;
        const int lane = r / 4;
        const int j    = r % 4;
        const int h = lane >> 4, lr = lane & 15;
        const int n  = nt * 16 + lr;
        const int kt = 2 * ktp + (j >> 1);
        const int k  = 4 * kt + 2 * h + (j & 1);
        val = (n < ACTION_NUM) ? outw[n * D_MODEL + k] : 0.0f;  // zero-pad 300..303
    }
    ws[idx] = val;
}

// ---------------- main kernel: 32 batch rows per wave, packed B fragments ----------------
__global__ __launch_bounds__(256)
void carnn_wmma_packed(const int*   __restrict__ actions,  // [B][9]
                       const float* __restrict__ emb,      // [301][64]
                       const float* __restrict__ Mb,       // [50][64]
                       const float* __restrict__ Wb,       // [56][64]
                       const float* __restrict__ outb,     // [300]
                       const float* __restrict__ wsf,      // packed fragments
                       float*       __restrict__ out,      // [B][300]
                       int batch)
{
    __shared__ float lds[8][2][16][68];    // two hl tiles per wave, padded stride

    const int lane = threadIdx.x & 31;
    const int wv   = threadIdx.x >> 5;
    const int half = lane >> 4;
    const int lr   = lane & 15;
    const int wave_id = blockIdx.x * 8 + wv;
    const int row0 = wave_id * 32;
    if (row0 >= batch) return;
    const int row1 = (row0 + 16 < batch) ? (row0 + 16) : row0;  // clamp (dup is harmless)

    float (*hl0)[68] = lds[wv][0];
    float (*hl1)[68] = lds[wv][1];
    const v4f* wsv = (const v4f*)wsf;      // v4f strides: s=1024, nt=256, ktp=32, lane=1

    v8f acc0[4], acc1[4];

    for (int s = 0; s < SEQ_LEN; ++s) {
        const int a0 = actions[(row0 + lr) * SEQ_LEN + s];
        const int a1 = actions[(row1 + lr) * SEQ_LEN + s];
        const float* xr0 = emb + a0 * D_MODEL;
        const float* xr1 = emb + a1 * D_MODEL;

        #pragma unroll
        for (int nt = 0; nt < 4; ++nt) { acc0[nt] = (v8f)0.0f; acc1[nt] = (v8f)0.0f; }

        // acc += x @ Mw[s].T
        {
            const v4f* bm = wsv + (WS_M_OFF / 4) + s * 1024;
            #pragma unroll 4
            for (int ktp = 0; ktp < 8; ++ktp) {
                const int ke = 8 * ktp + 2 * half;
                v2f a0e = *(const v2f*)(xr0 + ke);
                v2f a0o = *(const v2f*)(xr0 + ke + 4);
                v2f a1e = *(const v2f*)(xr1 + ke);
                v2f a1o = *(const v2f*)(xr1 + ke + 4);
                v4f b[4];
                #pragma unroll
                for (int nt = 0; nt < 4; ++nt) b[nt] = bm[nt * 256 + ktp * 32 + lane];
                #pragma unroll
                for (int nt = 0; nt < 4; ++nt) {
                    v2f be = {b[nt][0], b[nt][1]};
                    acc0[nt] = __builtin_amdgcn_wmma_f32_16x16x4_f32(false, a0e, false, be, (short)0, acc0[nt], false, false);
                    acc1[nt] = __builtin_amdgcn_wmma_f32_16x16x4_f32(false, a1e, false, be, (short)0, acc1[nt], false, false);
                }
                #pragma unroll
                for (int nt = 0; nt < 4; ++nt) {
                    v2f bo = {b[nt][2], b[nt][3]};
                    acc0[nt] = __builtin_amdgcn_wmma_f32_16x16x4_f32(false, a0o, false, bo, (short)0, acc0[nt], false, false);
                    acc1[nt] = __builtin_amdgcn_wmma_f32_16x16x4_f32(false, a1o, false, bo, (short)0, acc1[nt], false, false);
                }
            }
        }

        // acc += hl @ Ww[s].T   (hl == 0 at s == 0)
        if (s > 0) {
            const v4f* bw = wsv + (WS_W_OFF / 4) + s * 1024;
            #pragma unroll 4
            for (int ktp = 0; ktp < 8; ++ktp) {
                const int ke = 8 * ktp + 2 * half;
                v2f a0e = *(const v2f*)(&hl0[lr][ke]);
                v2f a0o = *(const v2f*)(&hl0[lr][ke + 4]);
                v2f a1e = *(const v2f*)(&hl1[lr][ke]);
                v2f a1o = *(const v2f*)(&hl1[lr][ke + 4]);
                v4f b[4];
                #pragma unroll
                for (int nt = 0; nt < 4; ++nt) b[nt] = bw[nt * 256 + ktp * 32 + lane];
                #pragma unroll
                for (int nt = 0; nt < 4; ++nt) {
                    v2f be = {b[nt][0], b[nt][1]};
                    acc0[nt] = __builtin_amdgcn_wmma_f32_16x16x4_f32(false, a0e, false, be, (short)0, acc0[nt], false, false);
                    acc1[nt] = __builtin_amdgcn_wmma_f32_16x16x4_f32(false, a1e, false, be, (short)0, acc1[nt], false, false);
                }
                #pragma unroll
                for (int nt = 0; nt < 4; ++nt) {
                    v2f bo = {b[nt][2], b[nt][3]};
                    acc0[nt] = __builtin_amdgcn_wmma_f32_16x16x4_f32(false, a0o, false, bo, (short)0, acc0[nt], false, false);
                    acc1[nt] = __builtin_amdgcn_wmma_f32_16x16x4_f32(false, a1o, false, bo, (short)0, acc1[nt], false, false);
                }
            }
        }

        // bias + sigmoid, write new hl tiles (D layout -> row-major LDS tile)
        #pragma unroll
        for (int nt = 0; nt < 4; ++nt) {
            const int n = nt * 16 + lr;
            const float bias = Mb[s * D_MODEL + n] + Wb[s * D_MODEL + n];
            #pragma unroll
            for (int r = 0; r < 8; ++r) {
                float v0 = sigmoid_f(acc0[nt][r] + bias);
                float v1 = sigmoid_f(acc1[nt][r] + bias);
                hl0[half * 8 + r][n] = v0;
                hl1[half * 8 + r][n] = v1;
            }
        }
        // wave-private LDS tiles: in-wave DS ordering suffices, no barriers
    }

    // ---- output projection: out = hl @ out_w.T + out_b ----
    const v4f* bo_base = wsv + (WS_O_OFF / 4);
    #pragma unroll 1
    for (int t = 0; t < 2; ++t) {
        if (t && row1 == row0) break;  // uniform tail guard
        float (*hlp)[68] = t ? hl1 : hl0;
        const int rbase = t ? row1 : row0;

        v2f afrag[16];
        #pragma unroll
        for (int kt = 0; kt < 16; ++kt)
            afrag[kt] = *(const v2f*)(&hlp[lr][4 * kt + 2 * half]);

        for (int nt0 = 0; nt0 < NT_OUT; nt0 += 4) {
            v8f oacc[4];
            #pragma unroll
            for (int j = 0; j < 4; ++j) oacc[j] = (v8f)0.0f;

            #pragma unroll 2
            for (int ktp = 0; ktp < 8; ++ktp) {
                #pragma unroll
                for (int j = 0; j < 4; ++j) {
                    const int nt = nt0 + j;
                    if (nt >= NT_OUT) break;               // uniform
                    v4f b = bo_base[nt * 256 + ktp * 32 + lane];
                    v2f be = {b[0], b[1]};
                    v2f bo = {b[2], b[3]};
                    oacc[j] = __builtin_amdgcn_wmma_f32_16x16x4_f32(false, afrag[2 * ktp],     false, be, (short)0, oacc[j], false, false);
                    oacc[j] = __builtin_amdgcn_wmma_f32_16x16x4_f32(false, afrag[2 * ktp + 1], false, bo, (short)0, oacc[j], false, false);
                }
            }

            #pragma unroll
            for (int j = 0; j < 4; ++j) {
                const int nt = nt0 + j;
                if (nt >= NT_OUT) break;                   // uniform
                const int n = nt * 16 + lr;
                if (n < ACTION_NUM) {
                    const float bias = outb[n];
                    #pragma unroll
                    for (int r = 0; r < 8; ++r) {
                        out[(long)(rbase + half * 8 + r) * ACTION_NUM + n] = oacc[j][r] + bias;
                    }
                }
            }
        }
    }
}

// ---------------- fallback (no workspace): round-1 direct-load kernel ----------------
__global__ __launch_bounds__(256)
void carnn_wmma_direct(const int*   __restrict__ actions,
                       const float* __restrict__ emb,
                       const float* __restrict__ Mw,
                       const float* __restrict__ Mb,
                       const float* __restrict__ Ww,
                       const float* __restrict__ Wb,
                       const float* __restrict__ outw,
                       const float* __restrict__ outb,
                       float*       __restrict__ out,
                       int batch)
{
    __shared__ float lds[8][16][68];
    const int lane = threadIdx.x & 31;
    const int wv   = threadIdx.x >> 5;
    const int half = lane >> 4;
    const int lr   = lane & 15;
    const int tile = blockIdx.x * 8 + wv;
    const int row0 = tile * 16;
    if (row0 >= batch) return;

    float (*hl)[68] = lds[wv];
    v8f acc[4];

    for (int s = 0; s < SEQ_LEN; ++s) {
        const float* mwp = Mw + s * D_MODEL * D_MODEL;
        const float* wwp = Ww + s * D_MODEL * D_MODEL;
        const int aidx = actions[(row0 + lr) * SEQ_LEN + s];
        const float* xrow = emb + aidx * D_MODEL;

        #pragma unroll
        for (int nt = 0; nt < 4; ++nt) acc[nt] = (v8f)0.0f;

        #pragma unroll 4
        for (int kt = 0; kt < 16; ++kt) {
            const int k = kt * 4 + half * 2;
            v2f a = *(const v2f*)(xrow + k);
            #pragma unroll
            for (int nt = 0; nt < 4; ++nt) {
                const int n = nt * 16 + lr;
                v2f b = *(const v2f*)(mwp + n * D_MODEL + k);
                acc[nt] = __builtin_amdgcn_wmma_f32_16x16x4_f32(false, a, false, b, (short)0, acc[nt], false, false);
            }
        }
        if (s > 0) {
            #pragma unroll 4
            for (int kt = 0; kt < 16; ++kt) {
                const int k = kt * 4 + half * 2;
                v2f a = *(const v2f*)(&hl[lr][k]);
                #pragma unroll
                for (int nt = 0; nt < 4; ++nt) {
                    const int n = nt * 16 + lr;
                    v2f b = *(const v2f*)(wwp + n * D_MODEL + k);
                    acc[nt] = __builtin_amdgcn_wmma_f32_16x16x4_f32(false, a, false, b, (short)0, acc[nt], false, false);
                }
            }
        }
        #pragma unroll
        for (int nt = 0; nt < 4; ++nt) {
            const int n = nt * 16 + lr;
            const float bias = Mb[s * D_MODEL + n] + Wb[s * D_MODEL + n];
            #pragma unroll
            for (int r = 0; r < 8; ++r) {
                float v = sigmoid_f(acc[nt][r] + bias);
                hl[half * 8 + r][n] = v;
            }
        }
    }

    v2f afrag[16];
    #pragma unroll
    for (int kt = 0; kt < 16; ++kt)
        afrag[kt] = *(const v2f*)(&hl[lr][4 * kt + 2 * half]);

    for (int nt0 = 0; nt0 < NT_OUT; nt0 += 4) {
        v8f oacc[4];
        #pragma unroll
        for (int j = 0; j < 4; ++j) oacc[j] = (v8f)0.0f;
        #pragma unroll 4
        for (int kt = 0; kt < 16; ++kt) {
            #pragma unroll
            for (int j = 0; j < 4; ++j) {
                const int nt = nt0 + j;
                if (nt >= NT_OUT) break;
                const int n = nt * 16 + lr;
                const int k = kt * 4 + half * 2;
                v2f b;
                if (n < ACTION_NUM) b = *(const v2f*)(outw + n * D_MODEL + k);
                else                b = (v2f)0.0f;
                oacc[j] = __builtin_amdgcn_wmma_f32_16x16x4_f32(false, afrag[kt], false, b, (short)0, oacc[j], false, false);
            }
        }
        #pragma unroll
        for (int j = 0; j < 4; ++j) {
            const int nt = nt0 + j;
            if (nt >= NT_OUT) break;
            const int n = nt * 16 + lr;
            if (n < ACTION_NUM) {
                const float bias = outb[n];
                #pragma unroll
                for (int r = 0; r < 8; ++r)
                    out[(long)(row0 + half * 8 + r) * ACTION_NUM + n] = oacc[j][r] + bias;
            }
        }
    }
}

extern "C" void kernel_launch(void* const* d_in, const int* in_sizes, int n_in,
                              void* d_out, int out_size, void* d_ws, size_t ws_size,
                              hipStream_t stream) {
    const int*   actions = (const int*)  d_in[0];
    const float* emb     = (const float*)d_in[1];
    const float* Mw      = (const float*)d_in[2];
    const float* Mb      = (const float*)d_in[3];
    const float* Ww      = (const float*)d_in[4];
    const float* Wb      = (const float*)d_in[5];
    const float* outw    = (const float*)d_in[6];
    const float* outb    = (const float*)d_in[7];
    float*       out     = (float*)d_out;

    const int batch = in_sizes[0] / SEQ_LEN;           // 65536
    const int tiles = (batch + 15) / 16;               // 4096

    if (ws_size >= (size_t)WS_BYTES && d_ws != nullptr) {
        float* ws = (float*)d_ws;
        carnn_pack_weights<<<(WS_FLOATS + 255) / 256, 256, 0, stream>>>(Mw, Ww, outw, ws);
        const int waves  = (tiles + 1) / 2;            // 32 rows per wave
        const int blocks = (waves + 7) / 8;            // 8 waves per block
        carnn_wmma_packed<<<blocks, 256, 0, stream>>>(actions, emb, Mb, Wb, outb, ws, out, batch);
    } else {
        const int blocks = (tiles + 7) / 8;
        carnn_wmma_direct<<<blocks, 256, 0, stream>>>(actions, emb, Mw, Mb, Ww, Wb, outw, outb, out, batch);
    }
}